// KineticPooling_73632919322743
// MI455X (gfx1250) — compile-verified
//
#include <hip/hip_runtime.h>

// CDNA5 / gfx1250: wave32, WMMA f32 16x16x4.
typedef __attribute__((ext_vector_type(2))) float v2f;
typedef __attribute__((ext_vector_type(8))) float v8f;

#define LDS_S 68   // row stride in floats: keeps float4 16B-aligned, few bank conflicts
#define NMAT 64

// C[16x16] -= A[16x16] * B[16x16], all tiles resident in LDS (row stride LDS_S).
// Implemented as 4 x V_WMMA_F32_16X16X4_F32 with the A fragment negated on load
// (f32 WMMA NEG bits only cover C, not A/B).
__device__ __forceinline__ void tile_gemm_sub(float* C, int rC, int cC,
                                              const float* A, int rA, int cA,
                                              const float* B, int rB, int cB,
                                              int lane) {
  const int half = lane >> 4;   // 0: lanes 0-15, 1: lanes 16-31
  const int l16  = lane & 15;
  // C/D layout: VGPR v holds M = v + 8*half, N = l16
  v8f c;
#pragma unroll
  for (int v = 0; v < 8; ++v)
    c[v] = C[(rC + half * 8 + v) * LDS_S + cC + l16];
#pragma unroll
  for (int kk = 0; kk < 16; kk += 4) {
    const int ka = kk + half * 2;   // A: lanes 0-15 hold K=kk,kk+1 ; lanes 16-31 hold K=kk+2,kk+3
    v2f a, b;
    a[0] = -A[(rA + l16) * LDS_S + cA + ka];
    a[1] = -A[(rA + l16) * LDS_S + cA + ka + 1];
    // B mirrors A: VGPR0 = rows K=kk (lanes 0-15) / K=kk+2 (lanes 16-31), VGPR1 = +1
    b[0] = B[(rB + ka)     * LDS_S + cB + l16];
    b[1] = B[(rB + ka + 1) * LDS_S + cB + l16];
    c = __builtin_amdgcn_wmma_f32_16x16x4_f32(false, a, false, b, (short)0, c, false, false);
  }
#pragma unroll
  for (int v = 0; v < 8; ++v)
    C[(rC + half * 8 + v) * LDS_S + cC + l16] = c[v];
}

__global__ __launch_bounds__(256) void kinetic_lu_trace_kernel(
    const float* __restrict__ Aup, const float* __restrict__ Adown,
    const float* __restrict__ d2Aup, const float* __restrict__ d2Adown,
    const int* __restrict__ flag, float* __restrict__ out) {
  __shared__ float As[NMAT * LDS_S];   // 17 KB: A, then LU factors in place
  __shared__ float Bs[NMAT * LDS_S];   // 17 KB: B -> Y -> X in place
  __shared__ float red[NMAT];
  __shared__ int   sPiv;
  __shared__ int   sSwaps;
  __shared__ float sTrace;
  __shared__ float sDet;

  const int t    = threadIdx.x;   // 256 threads = 8 wave32
  const int wave = t >> 5;
  const int lane = t & 31;
  const int bc   = blockIdx.x;    // (batch, conf) flattened, row-major like the reference
  const size_t moff = (size_t)bc * (NMAT * NMAT);
  const int le = flag[0];         // return_local_energy (uniform scalar)

  if (t == 0) { sTrace = 0.f; sDet = 1.f; }

  for (int spin = 0; spin < 2; ++spin) {
    const float* Ag = spin ? Adown   : Aup;
    const float* Bg = spin ? d2Adown : d2Aup;
    if (t == 0) sSwaps = 0;

    // ---- load A, B into LDS (float4, coalesced, single pass over HBM) ----
    const float4* A4 = reinterpret_cast<const float4*>(Ag + moff);
    const float4* B4 = reinterpret_cast<const float4*>(Bg + moff);
    for (int i4 = t; i4 < NMAT * NMAT / 4; i4 += 256) {
      const int r = i4 >> 4;          // 16 float4 per 64-wide row
      const int c = (i4 & 15) << 2;
      float4 va = A4[i4];
      float4 vb = B4[i4];
      float* pa = &As[r * LDS_S + c];
      float* pb = &Bs[r * LDS_S + c];
      pa[0] = va.x; pa[1] = va.y; pa[2] = va.z; pa[3] = va.w;
      pb[0] = vb.x; pb[1] = vb.y; pb[2] = vb.z; pb[3] = vb.w;
    }
    __syncthreads();

    // ---- blocked LU with partial pivoting (NB = 16) ----
    for (int kb = 0; kb < 4; ++kb) {
      const int c0 = kb * 16;
      for (int j = c0; j < c0 + 16; ++j) {
        // pivot search on column j (wave 0, shuffle argmax)
        if (wave == 0) {
          float best = -1.f; int bi = j;
          for (int i = j + lane; i < NMAT; i += 32) {
            float v = fabsf(As[i * LDS_S + j]);
            if (v > best) { best = v; bi = i; }
          }
          for (int off = 16; off > 0; off >>= 1) {
            float ob = __shfl_xor(best, off);
            int   oi = __shfl_xor(bi, off);
            if (ob > best) { best = ob; bi = oi; }
          }
          if (lane == 0) { sPiv = bi; if (bi != j) sSwaps++; }
        }
        __syncthreads();
        const int p = sPiv;           // block-uniform
        if (p != j) {
          if (t < NMAT) {
            float tmp = As[j * LDS_S + t];
            As[j * LDS_S + t] = As[p * LDS_S + t];
            As[p * LDS_S + t] = tmp;
          } else if (t < 2 * NMAT) {
            const int c = t - NMAT;
            float tmp = Bs[j * LDS_S + c];
            Bs[j * LDS_S + c] = Bs[p * LDS_S + c];
            Bs[p * LDS_S + c] = tmp;
          }
          __syncthreads();
        }
        // multipliers + rank-1 update restricted to panel columns
        const int i = j + 1 + t;
        if (i < NMAT) {
          const float inv = 1.0f / As[j * LDS_S + j];
          float l = As[i * LDS_S + j] * inv;
          As[i * LDS_S + j] = l;
          for (int jj = j + 1; jj < c0 + 16; ++jj)
            As[i * LDS_S + jj] -= l * As[j * LDS_S + jj];
        }
        __syncthreads();
      }
      const int ncols = 48 - c0;      // trailing width
      if (ncols > 0) {
        // TRSM: U12 = L11^{-1} A12  (per-column forward substitution)
        if (t < ncols) {
          const int cc = c0 + 16 + t;
          for (int r = 1; r < 16; ++r) {
            float x = As[(c0 + r) * LDS_S + cc];
            for (int q = 0; q < r; ++q)
              x -= As[(c0 + r) * LDS_S + (c0 + q)] * As[(c0 + q) * LDS_S + cc];
            As[(c0 + r) * LDS_S + cc] = x;
          }
        }
        __syncthreads();
        // trailing update A22 -= L21 * U12  (WMMA tiles, one tile per wave)
        const int nb = 3 - kb;
        const int ntiles = nb * nb;
        for (int tile = wave; tile < ntiles; tile += 8) {
          const int ti = tile / nb;
          const int tj = tile % nb;
          const int r0  = c0 + 16 + ti * 16;
          const int cc0 = c0 + 16 + tj * 16;
          tile_gemm_sub(As, r0, cc0, As, r0, c0, As, c0, cc0, lane);
        }
        __syncthreads();
      }
    }

    // ---- forward solve  L Y = P B  (unit lower, 64 RHS, in-place in Bs) ----
    for (int ib = 0; ib < 4; ++ib) {
      const int ntiles = ib * 4;      // B_ib -= sum_{jb<ib} L[ib][jb] * Y[jb]
      for (int tile = wave; tile < ntiles; tile += 8) {
        const int jb = tile >> 2;
        const int cb = tile & 3;
        tile_gemm_sub(Bs, ib * 16, cb * 16, As, ib * 16, jb * 16, Bs, jb * 16, cb * 16, lane);
      }
      __syncthreads();
      if (t < NMAT) {                 // 16x16 unit-lower diagonal solve, one column per thread
        const int base = ib * 16;
        for (int r = 1; r < 16; ++r) {
          float x = Bs[(base + r) * LDS_S + t];
          for (int q = 0; q < r; ++q)
            x -= As[(base + r) * LDS_S + (base + q)] * Bs[(base + q) * LDS_S + t];
          Bs[(base + r) * LDS_S + t] = x;
        }
      }
      __syncthreads();
    }

    // ---- back solve  U X = Y ----
    for (int ib = 3; ib >= 0; --ib) {
      const int ntiles = (3 - ib) * 4;
      for (int tile = wave; tile < ntiles; tile += 8) {
        const int jb = ib + 1 + (tile >> 2);
        const int cb = tile & 3;
        tile_gemm_sub(Bs, ib * 16, cb * 16, As, ib * 16, jb * 16, Bs, jb * 16, cb * 16, lane);
      }
      __syncthreads();
      if (t < NMAT) {                 // 16x16 upper diagonal solve
        const int base = ib * 16;
        for (int r = 15; r >= 0; --r) {
          float x = Bs[(base + r) * LDS_S + t];
          for (int q = r + 1; q < 16; ++q)
            x -= As[(base + r) * LDS_S + (base + q)] * Bs[(base + q) * LDS_S + t];
          Bs[(base + r) * LDS_S + t] = x / As[(base + r) * LDS_S + (base + r)];
        }
      }
      __syncthreads();
    }

    // ---- trace(X) (+ optional det from U diagonal & pivot parity) ----
    if (t < NMAT) red[t] = Bs[t * LDS_S + t];
    __syncthreads();
    if (wave == 0) {
      float s = red[lane] + red[lane + 32];
      for (int off = 16; off > 0; off >>= 1) s += __shfl_xor(s, off);
      if (le == 0) {
        float pdet = 1.f;
        for (int i = lane; i < NMAT; i += 32) pdet *= As[i * LDS_S + i];
        for (int off = 16; off > 0; off >>= 1) pdet *= __shfl_xor(pdet, off);
        if (lane == 0) { if (sSwaps & 1) pdet = -pdet; sDet *= pdet; }
      }
      if (lane == 0) sTrace += s;
    }
    __syncthreads();
  }

  if (t == 0) {
    float r = -0.5f * sTrace;
    if (le == 0) r *= sDet;
    out[bc] = r;
  }
}

extern "C" void kernel_launch(void* const* d_in, const int* in_sizes, int n_in,
                              void* d_out, int out_size, void* d_ws, size_t ws_size,
                              hipStream_t stream) {
  const float* Aup     = (const float*)d_in[0];
  const float* Adown   = (const float*)d_in[1];
  const float* d2Aup   = (const float*)d_in[2];
  const float* d2Adown = (const float*)d_in[3];
  const int*   flag    = (const int*)d_in[4];
  float* out = (float*)d_out;

  const int nprob = in_sizes[0] / (NMAT * NMAT);   // Nbatch * Nconfs = 8192
  hipLaunchKernelGGL(kinetic_lu_trace_kernel, dim3(nprob), dim3(256), 0, stream,
                     Aup, Adown, d2Aup, d2Adown, flag, out);
}